// DotPredictor_67972152426915
// MI455X (gfx1250) — compile-verified
//
#include <hip/hip_runtime.h>

// u_dot_v: score[e] = dot(h[src[e]], h[dst[e]]), d = 128, f32, E = 600000.
//
// Roofline: 614 MB of gathered rows for 0.154 GFLOP (0.25 FLOP/byte) -> purely
// memory-bound, WMMA cannot contribute. h (51.2 MB) is L2-resident (192 MB L2).
//
// CDNA5-native path: the Tensor Data Mover in GATHER MODE DMAs 8 indexed rows
// of h into LDS per instruction (32-bit row indices in D# groups 2/3). Per
// 8-edge tile a single-wave workgroup issues 2 tensor_load_to_lds ops (hu
// rows, hv rows), waits on TENSORcnt, then does conflict-free ds_load_b128
// reads + FMA + 8-lane ds_swizzle butterfly. Descriptor padding
// (pad_interval=128dw, pad_amount=16dw -> 144-dword edge stride) puts the four
// 8-lane groups at banks 0/16/32/48 so every LDS b128 read hits each bank
// exactly twice (the minimum).
//
// Single-wave workgroups: index loads are uniform -> pure s_load_b256 (no
// vector-memory traffic for addresses), LDS is private per wave (~9 KB), and
// 32 workgroups/WGP (8 waves/SIMD) hide the DMA latency across waves.
//
// Tail handling: the last tile start is clamped to n_edges-8; overlapping
// tiles recompute identical values (idempotent stores), so there is a single
// straight-line code path. Requires n_edges >= 8 (harness: E = 600000).
//
// NOTE: this toolchain exposes the 6-arg tensor_load_to_lds builtin
// (v4u g0, v8i g1, v4i g2, v4i g3, v8i extra, i32 cpol); the ISA D# has only
// four groups, so the extra operand is passed zero-filled.

typedef unsigned int v4u __attribute__((ext_vector_type(4)));
typedef int          v8i __attribute__((ext_vector_type(8)));
typedef int          v4i __attribute__((ext_vector_type(4)));

#define D_FEAT          128
#define EDGE_STRIDE_DW  144   // 128 dw row + 16 dw TDM pad (bank-conflict-free)

template <int PATTERN>
__device__ __forceinline__ float swz_add(float v) {
  int o = __builtin_amdgcn_ds_swizzle(__float_as_int(v), PATTERN);
  return v + __int_as_float(o);
}

__device__ __forceinline__ int rfl(int x) { return __builtin_amdgcn_readfirstlane(x); }

// TDM gather-mode descriptor: DMA 8 rows (h[idx[i]][0..127], f32) into LDS at
// lds_off_bytes, with +16dw padding after each 128dw row.
__device__ __forceinline__ void tdm_gather_rows8(unsigned lds_off_bytes,
                                                 const float* h, int n_nodes,
                                                 v4i idx_lo, v4i idx_hi) {
  unsigned long long ga = (unsigned long long)(uintptr_t)h;
  v4u g0;
  g0[0] = 0xC0000001u;                       // count=1 | gather_index_size=32b | gather_mode=1
  g0[1] = lds_off_bytes;                     // lds_addr
  g0[2] = (unsigned)ga;                      // global_addr[31:0]
  g0[3] = ((unsigned)(ga >> 32) & 0x01FFFFFFu) | (2u << 30);  // global_addr[56:32] | type=2
  v8i g1;
  g1[0] = (2 << 16)            // data_size = 4 bytes
        | (1 << 20)            // pad_enable
        | (6 << 22)            // pad_interval: 2^(6+1) = 128 dwords (one row)
        | (15 << 25);          // pad_amount: 15+1 = 16 dwords
  g1[1] = (D_FEAT & 0xFFFF) << 16;                       // tensor_dim0[15:0]
  g1[2] = (n_nodes & 0xFFFF) << 16;                      // dim0[31:16]=0 | tensor_dim1[15:0]
  g1[3] = ((n_nodes >> 16) & 0xFFFF) | (D_FEAT << 16);   // tensor_dim1[31:16] | tile_dim0
  g1[4] = 8;                                             // tile_dim1 = #valid indices
  g1[5] = D_FEAT;                                        // tensor_dim0_stride[31:0]
  g1[6] = 0;
  g1[7] = 0;
  v8i gx = {0, 0, 0, 0, 0, 0, 0, 0};                     // undocumented 5th operand: zero
  __builtin_amdgcn_tensor_load_to_lds(g0, g1, idx_lo, idx_hi, gx, 0);
}

__global__ __launch_bounds__(32)
void DotPredictor_u_dot_v_tdm(const float* __restrict__ h,
                              const int* __restrict__ src,
                              const int* __restrict__ dst,
                              float* __restrict__ out,
                              int n_edges, int n_nodes) {
  __shared__ float lds_u[8 * EDGE_STRIDE_DW];
  __shared__ float lds_v[8 * EDGE_STRIDE_DW];

  const int lane = (int)(threadIdx.x & 31u);
  const int grp  = lane >> 3;        // 0..3 : edge-group within wave
  const int j    = lane & 7;         // 0..7 : lane within group

  // Tile start, clamped so the last tile overlaps instead of branching.
  int e0 = (int)blockIdx.x * 8;
  if (e0 + 8 > n_edges) e0 = n_edges - 8;   // n_edges >= 8 assumed

  // ---- 16 uniform index loads -> SMEM (s_load), pinned to SGPRs ----
  v4i silo, sihi, dilo, dihi;
  #pragma unroll
  for (int t = 0; t < 4; ++t) {
    silo[t] = rfl(src[e0 + t]);
    sihi[t] = rfl(src[e0 + 4 + t]);
    dilo[t] = rfl(dst[e0 + t]);
    dihi[t] = rfl(dst[e0 + 4 + t]);
  }

  // ---- DMA-gather 8 hu rows and 8 hv rows into LDS (async, TENSORcnt) ----
  tdm_gather_rows8((unsigned)(uintptr_t)&lds_u[0], h, n_nodes, silo, sihi);
  tdm_gather_rows8((unsigned)(uintptr_t)&lds_v[0], h, n_nodes, dilo, dihi);
  __builtin_amdgcn_s_wait_tensorcnt(0);

  // ---- two passes of 4 edges: 8 lanes per edge, conflict-free b128 reads ----
  #pragma unroll
  for (int p = 0; p < 2; ++p) {
    const int el = p * 4 + grp;                     // edge-in-tile 0..7
    const float* bu = lds_u + el * EDGE_STRIDE_DW + (j << 2);
    const float* bv = lds_v + el * EDGE_STRIDE_DW + (j << 2);

    float4 u0 = *(const float4*)(bu);
    float4 u1 = *(const float4*)(bu + 32);
    float4 u2 = *(const float4*)(bu + 64);
    float4 u3 = *(const float4*)(bu + 96);
    float4 v0 = *(const float4*)(bv);
    float4 v1 = *(const float4*)(bv + 32);
    float4 v2 = *(const float4*)(bv + 64);
    float4 v3 = *(const float4*)(bv + 96);

    float ax = u0.x * v0.x;
    float ay = u0.y * v0.y;
    float az = u0.z * v0.z;
    float aw = u0.w * v0.w;
    ax = fmaf(u1.x, v1.x, ax);
    ay = fmaf(u1.y, v1.y, ay);
    az = fmaf(u1.z, v1.z, az);
    aw = fmaf(u1.w, v1.w, aw);
    ax = fmaf(u2.x, v2.x, ax);
    ay = fmaf(u2.y, v2.y, ay);
    az = fmaf(u2.z, v2.z, az);
    aw = fmaf(u2.w, v2.w, aw);
    ax = fmaf(u3.x, v3.x, ax);
    ay = fmaf(u3.y, v3.y, ay);
    az = fmaf(u3.z, v3.z, az);
    aw = fmaf(u3.w, v3.w, aw);

    float s = (ax + ay) + (az + aw);
    s = swz_add<0x041F>(s);   // xor 1
    s = swz_add<0x081F>(s);   // xor 2
    s = swz_add<0x101F>(s);   // xor 4

    if (j == 0) {
      __builtin_nontemporal_store(s, out + e0 + el);  // stream out, keep h in L2
    }
  }
}

extern "C" void kernel_launch(void* const* d_in, const int* in_sizes, int n_in,
                              void* d_out, int out_size, void* d_ws, size_t ws_size,
                              hipStream_t stream) {
  const float* h   = (const float*)d_in[0];
  const int*   src = (const int*)d_in[1];
  const int*   dst = (const int*)d_in[2];
  float*       out = (float*)d_out;

  const int n_edges = in_sizes[1];
  const int n_nodes = in_sizes[0] / D_FEAT;
  const int blocks  = (n_edges + 7) / 8;     // one 8-edge tile per single-wave WG

  hipLaunchKernelGGL(DotPredictor_u_dot_v_tdm,
                     dim3(blocks), dim3(32), 0, stream,
                     h, src, dst, out, n_edges, n_nodes);
}